// MHCLayer_86964497809924
// MI455X (gfx1250) — compile-verified
//
#include <hip/hip_runtime.h>
#include <math.h>

// Problem geometry (fixed by the reference): x (4,4096,4096), E=4, D=4096.
#define D 4096
#define E 4
#define NTOK 16384          // B*S = 4*4096 tokens
#define SINK_ITERS 20
#define SINK_EPS 1e-6f
#define TPG 16              // tokens per wave group (WMMA M dimension)
#define WAVES 4             // waves per block
#define BLOCK (WAVES * 32)

typedef __attribute__((ext_vector_type(2))) float v2f;
typedef __attribute__((ext_vector_type(8))) float v8f;
typedef __attribute__((ext_vector_type(4))) int   v4i;

// Async copy builtin takes typed int4 pointers: (addrspace(1), addrspace(3)).
typedef __attribute__((address_space(1))) v4i* g4p;   // global int4*
typedef __attribute__((address_space(3))) v4i* l4p;   // LDS int4*

__device__ __forceinline__ float wave_sum(float v) {
#pragma unroll
    for (int off = 16; off > 0; off >>= 1) v += __shfl_xor(v, off, 32);
    return v;
}
__device__ __forceinline__ float wave_max(float v) {
#pragma unroll
    for (int off = 16; off > 0; off >>= 1) v = fmaxf(v, __shfl_xor(v, off, 32));
    return v;
}

// ---------------------------------------------------------------------------
// Kernel 1: tiny prep. sinkhorn(H_res 4x4), softmax rows of H_pre/H_post,
// fold H_res into H_post:  H_post_eff = H_res @ softmax(H_post_logits).
// One block, 128 threads (4 waves; wave w owns row w of each 4xD table).
// ---------------------------------------------------------------------------
__global__ void __launch_bounds__(128) mhc_prepare(
        const float* __restrict__ hres_logits,
        const float* __restrict__ hpre_logits,
        const float* __restrict__ hpost_logits,
        float* __restrict__ ws_pre,        // (E,D) softmax(H_pre)
        float* __restrict__ ws_posteff) {  // (E,D) H_res @ softmax(H_post)
    __shared__ float s_post[E * D];        // 64 KB
    __shared__ float s_hres[E * E];

    const int tid  = threadIdx.x;
    const int wave = tid >> 5;             // 0..3 == expert row
    const int lane = tid & 31;

    // --- row softmax of H_pre row `wave` -> ws_pre ---
    {
        const float* row = hpre_logits + wave * D;
        float m = -INFINITY;
        for (int d = lane; d < D; d += 32) m = fmaxf(m, row[d]);
        m = wave_max(m);
        float s = 0.f;
        for (int d = lane; d < D; d += 32) s += __expf(row[d] - m);
        s = wave_sum(s);
        const float inv = 1.f / s;
        for (int d = lane; d < D; d += 32)
            ws_pre[wave * D + d] = __expf(row[d] - m) * inv;
    }
    // --- row softmax of H_post row `wave` -> LDS ---
    {
        const float* row = hpost_logits + wave * D;
        float m = -INFINITY;
        for (int d = lane; d < D; d += 32) m = fmaxf(m, row[d]);
        m = wave_max(m);
        float s = 0.f;
        for (int d = lane; d < D; d += 32) s += __expf(row[d] - m);
        s = wave_sum(s);
        const float inv = 1.f / s;
        for (int d = lane; d < D; d += 32)
            s_post[wave * D + d] = __expf(row[d] - m) * inv;
    }
    // --- sinkhorn on the 4x4 (serial, thread 0) ---
    if (tid == 0) {
        float A[E * E];
        float m = hres_logits[0];
#pragma unroll
        for (int i = 1; i < E * E; ++i) m = fmaxf(m, hres_logits[i]);
#pragma unroll
        for (int i = 0; i < E * E; ++i) A[i] = __expf(hres_logits[i] - m);
        for (int it = 0; it < SINK_ITERS; ++it) {
#pragma unroll
            for (int r = 0; r < E; ++r) {
                float s = A[r*E+0] + A[r*E+1] + A[r*E+2] + A[r*E+3] + SINK_EPS;
                const float inv = 1.f / s;
#pragma unroll
                for (int c = 0; c < E; ++c) A[r*E+c] *= inv;
            }
#pragma unroll
            for (int c = 0; c < E; ++c) {
                float s = A[0*E+c] + A[1*E+c] + A[2*E+c] + A[3*E+c] + SINK_EPS;
                const float inv = 1.f / s;
#pragma unroll
                for (int r = 0; r < E; ++r) A[r*E+c] *= inv;
            }
        }
#pragma unroll
        for (int i = 0; i < E * E; ++i) s_hres[i] = A[i];
    }
    __syncthreads();

    // --- fold: ws_posteff[e][d] = sum_e' H_res[e][e'] * softmax(H_post)[e'][d]
    for (int d = tid; d < D; d += 128) {
        const float p0 = s_post[0*D + d], p1 = s_post[1*D + d];
        const float p2 = s_post[2*D + d], p3 = s_post[3*D + d];
#pragma unroll
        for (int e = 0; e < E; ++e) {
            ws_posteff[e * D + d] = s_hres[e*E+0]*p0 + s_hres[e*E+1]*p1
                                  + s_hres[e*E+2]*p2 + s_hres[e*E+3]*p3;
        }
    }
}

// ---------------------------------------------------------------------------
// Kernel 2: main streaming pass. Each wave owns 16 tokens.
//   Stage:   routing tables global -> LDS via GLOBAL_LOAD_ASYNC_TO_LDS_B128
//            (ASYNCcnt-tracked, no VGPR round trip).
//   Phase A: x_pre[t][e] = dot(x[t,:], H_pre[e,:])    (coalesced, LDS tables)
//   Phase B: out[t, d0:d0+16] via V_WMMA_F32_16X16X4_F32:
//            A = x_pre (16x4), B = H_post_eff[:, d0:d0+16] (4x16), K = 4 exact.
// ---------------------------------------------------------------------------
__global__ void __launch_bounds__(BLOCK) mhc_main(
        const float* __restrict__ x,
        const float* __restrict__ ws_pre,
        const float* __restrict__ ws_posteff,
        float* __restrict__ out) {
    __shared__ __align__(16) float s_pre[E * D];    // 64 KB
    __shared__ __align__(16) float s_post[E * D];   // 64 KB
    __shared__ float s_xpre[WAVES][TPG * E];        // per-wave x_pre tiles

    const int tid  = threadIdx.x;
    const int wave = tid >> 5;
    const int lane = tid & 31;

    // Stage routing tables into LDS with async cache->LDS b128 copies.
    // Each thread owns 16-byte chunks; 32 lanes cover 512B contiguous/issue.
    {
        const float4* gp = (const float4*)ws_pre;
        const float4* gq = (const float4*)ws_posteff;
        float4* sp = (float4*)s_pre;
        float4* sq = (float4*)s_post;
#pragma unroll 4
        for (int i = tid; i < (E * D) / 4; i += BLOCK) {
            __builtin_amdgcn_global_load_async_to_lds_b128(
                (g4p)(gp + i), (l4p)(sp + i), 0, 0);
            __builtin_amdgcn_global_load_async_to_lds_b128(
                (g4p)(gq + i), (l4p)(sq + i), 0, 0);
        }
        __builtin_amdgcn_s_wait_asynccnt(0);
    }
    __syncthreads();

    const int group = blockIdx.x * WAVES + wave;    // 0..1023
    const int tok0  = group * TPG;

    // ---------------- Phase A: x_pre via wave-cooperative dot products ------
    for (int t = 0; t < TPG; ++t) {
        const float* xrow = x + (size_t)(tok0 + t) * D;
        if (t + 1 < TPG)
            __builtin_prefetch(x + (size_t)(tok0 + t + 1) * D + lane * 128, 0, 1);
        float a0 = 0.f, a1 = 0.f, a2 = 0.f, a3 = 0.f;
#pragma unroll 4
        for (int it = 0; it < D / 128; ++it) {      // 32 lanes * float4
            const int d = (it * 32 + lane) * 4;
            const float4 xv = *(const float4*)(xrow + d);
            const float4 h0 = *(const float4*)(s_pre + 0 * D + d);
            const float4 h1 = *(const float4*)(s_pre + 1 * D + d);
            const float4 h2 = *(const float4*)(s_pre + 2 * D + d);
            const float4 h3 = *(const float4*)(s_pre + 3 * D + d);
            a0 += xv.x*h0.x + xv.y*h0.y + xv.z*h0.z + xv.w*h0.w;
            a1 += xv.x*h1.x + xv.y*h1.y + xv.z*h1.z + xv.w*h1.w;
            a2 += xv.x*h2.x + xv.y*h2.y + xv.z*h2.z + xv.w*h2.w;
            a3 += xv.x*h3.x + xv.y*h3.y + xv.z*h3.z + xv.w*h3.w;
        }
        a0 = wave_sum(a0); a1 = wave_sum(a1);
        a2 = wave_sum(a2); a3 = wave_sum(a3);
        if (lane == 0) {
            s_xpre[wave][t * E + 0] = a0;
            s_xpre[wave][t * E + 1] = a1;
            s_xpre[wave][t * E + 2] = a2;
            s_xpre[wave][t * E + 3] = a3;
        }
    }
    __syncthreads();

    // ---------------- Phase B: expand via WMMA f32 16x16x4 ------------------
    // A layout (16x4 f32): lanes 0-15 hold K=0 (v0), K=1 (v1) for M=lane;
    //                      lanes 16-31 hold K=2 (v0), K=3 (v1) for M=lane-16.
    const int row   = lane & 15;
    const int khalf = lane >> 4;   // 0 -> K{0,1}, 1 -> K{2,3}
    v2f a;
    a.x = s_xpre[wave][row * E + khalf * 2 + 0];
    a.y = s_xpre[wave][row * E + khalf * 2 + 1];

    for (int chunk = 0; chunk < D / 16; ++chunk) {
        const int d0 = chunk * 16;
        // B (4x16): same K split across lane halves, N = lane&15.
        v2f b;
        b.x = s_post[(khalf * 2 + 0) * D + d0 + row];
        b.y = s_post[(khalf * 2 + 1) * D + d0 + row];
        v8f c = {};
        c = __builtin_amdgcn_wmma_f32_16x16x4_f32(
                /*neg_a=*/false, a, /*neg_b=*/false, b,
                /*c_mod=*/(short)0, c, /*reuse_a=*/false, /*reuse_b=*/false);
        // C layout: VGPR r -> M = r (lanes 0-15) or r+8 (lanes 16-31), N = lane&15.
        const int tbase = tok0 + khalf * 8;
#pragma unroll
        for (int r = 0; r < 8; ++r) {
            out[(size_t)(tbase + r) * D + d0 + row] = c[r];
        }
    }
}

// ---------------------------------------------------------------------------
extern "C" void kernel_launch(void* const* d_in, const int* in_sizes, int n_in,
                              void* d_out, int out_size, void* d_ws, size_t ws_size,
                              hipStream_t stream) {
    const float* x     = (const float*)d_in[0];   // (4,4096,4096)
    const float* hres  = (const float*)d_in[1];   // (4,4)
    const float* hpre  = (const float*)d_in[2];   // (4,4096)
    const float* hpost = (const float*)d_in[3];   // (4,4096)
    float* out = (float*)d_out;                   // (4,4096,4096)

    float* ws_pre     = (float*)d_ws;             // E*D floats
    float* ws_posteff = ws_pre + E * D;           // E*D floats (128 KB total)

    hipLaunchKernelGGL(mhc_prepare, dim3(1), dim3(128), 0, stream,
                       hres, hpre, hpost, ws_pre, ws_posteff);

    const int groups = NTOK / TPG;                // 1024
    hipLaunchKernelGGL(mhc_main, dim3(groups / WAVES), dim3(BLOCK), 0, stream,
                       x, ws_pre, ws_posteff, out);
}